// NITS_20409684590837
// MI455X (gfx1250) — compile-verified
//
#include <hip/hip_runtime.h>

// ---------------------------------------------------------------------------
// NITS inverse-CDF (bisection) for ARCH=(1,32,32,1) on gfx1250.
//
// Strategy: one sample per wave32. Keep E = exp(-A1) (32x32 f32) resident in
// VGPRs as V_WMMA_F32_16X16X4_F32 A-fragments. Replace the 10-step bisection
// by an exactly-equivalent two-level 32-way section search over the same
// 2^-10 grid: each level evaluates f(x) at 32 probe points with a per-sample
// 32x32 @ 32x32 f32 GEMM on the WMMA pipe (2 Mtiles x 2 Ntiles x 8 Ksteps).
// All parameters are read exactly once (memory-bound: ~1.21 GB @ 23.3 TB/s).
// Cross-lane traffic uses ds_bpermute only (no LDS alloc, EXEC stays full).
// ---------------------------------------------------------------------------

typedef float v2f __attribute__((ext_vector_type(2)));
typedef float v8f __attribute__((ext_vector_type(8)));

#define NITS_MONO 0.001f

__device__ __forceinline__ float lane_gather(float v, int srcLane) {
  return __int_as_float(__builtin_amdgcn_ds_bpermute(srcLane << 2, __float_as_int(v)));
}
__device__ __forceinline__ int lane_gather_i(int v, int srcLane) {
  return __builtin_amdgcn_ds_bpermute(srcLane << 2, v);
}
__device__ __forceinline__ float wave_sum32(float v, int lane) {
#pragma unroll
  for (int m = 16; m; m >>= 1) v += lane_gather(v, lane ^ m);
  return v;
}
__device__ __forceinline__ float wave_max32(float v, int lane) {
#pragma unroll
  for (int m = 16; m; m >>= 1) v = fmaxf(v, lane_gather(v, lane ^ m));
  return v;
}
__device__ __forceinline__ int wave_maxi32(int v, int lane) {
#pragma unroll
  for (int m = 16; m; m >>= 1) {
    int o = lane_gather_i(v, lane ^ m);
    v = (o > v) ? o : v;
  }
  return v;
}

__device__ __forceinline__ float fexp(float x) { return __expf(x); }              // v_exp_f32
__device__ __forceinline__ float frcp(float x) { return __builtin_amdgcn_rcpf(x); } // v_rcp_f32
__device__ __forceinline__ float fsigmoid(float x) { return frcp(1.0f + fexp(-x)); }

#define PARAMS_PER 1152   // 32 (A0) + 32 (b0) + 1024 (A1) + 32 (b1) + 32 (A2)
#define WAVES_PER_BLOCK 8

__global__ __launch_bounds__(32 * WAVES_PER_BLOCK)
void NITS_invcdf_wmma_kernel(const float* __restrict__ z,
                             const float* __restrict__ params,
                             float* __restrict__ out, int n) {
  const int lane   = (int)(threadIdx.x & 31u);
  const int wid    = (int)(threadIdx.x >> 5);
  const int sample = (int)blockIdx.x * WAVES_PER_BLOCK + wid;
  if (sample >= n) return;                  // wave-uniform; EXEC stays full

  const float* p = params + (size_t)sample * PARAMS_PER;
  const int s = lane >> 4;                  // lane half (selects K pair / row half)
  const int r = lane & 15;                  // row-in-tile / probe-column id

  // ---- E = exp(-A1) as WMMA A-fragments (layout: lane = M + 16*(Klocal/2),
  //      vgpr = Klocal&1) + per-row sums for the layer-1 bias term.
  v2f ea[2][8];
  float rs0 = 0.0f, rs1 = 0.0f;
#pragma unroll
  for (int j = 0; j < 8; ++j) {
    v2f raw0 = *(const v2f*)(p + 64 + r * 32 + 4 * j + 2 * s);
    v2f raw1 = *(const v2f*)(p + 64 + (16 + r) * 32 + 4 * j + 2 * s);
    v2f e0, e1;
    e0.x = fexp(-raw0.x); e0.y = fexp(-raw0.y);
    e1.x = fexp(-raw1.x); e1.y = fexp(-raw1.y);
    ea[0][j] = e0; ea[1][j] = e1;
    rs0 += e0.x + e0.y; rs1 += e1.x + e1.y;
  }
  rs0 += lane_gather(rs0, lane ^ 16);       // complete row sums across K halves
  rs1 += lane_gather(rs1, lane ^ 16);
  const float c1m0 = p[1088 + r]      * rs0 * (1.0f / 32.0f);  // b1[o]*mean_i E[o,i]
  const float c1m1 = p[1088 + 16 + r] * rs1 * (1.0f / 32.0f);

  // ---- layer-0 params needed by this lane's B-fragment slots (i = 4j+2s+{0,1})
  float a0g[8][2], b0g[8][2];
#pragma unroll
  for (int j = 0; j < 8; ++j) {
    v2f a = *(const v2f*)(p + 4 * j + 2 * s);
    v2f b = *(const v2f*)(p + 32 + 4 * j + 2 * s);
    a0g[j][0] = a.x; a0g[j][1] = a.y;
    b0g[j][0] = b.x; b0g[j][1] = b.y;
  }

  // ---- w2 = softmax(A2) (lane i holds w2[i])
  const float a2 = p[1120 + lane];
  const float mx = wave_max32(a2, lane);
  const float ex = fexp(a2 - mx);
  const float w2l = ex * frcp(wave_sum32(ex, lane));

  // lane i holds c1[i]
  const float c1l = s ? c1m1 : c1m0;

  // ---- gather c1/w2 for this lane's D-fragment rows (row = 16m + 8s + v)
  float c1g[2][8], w2g[2][8];
#pragma unroll
  for (int m = 0; m < 2; ++m)
#pragma unroll
    for (int v = 0; v < 8; ++v) {
      const int rr = 16 * m + 8 * s + v;
      c1g[m][v] = lane_gather(c1l, rr);
      w2g[m][v] = lane_gather(w2l, rr);
    }

  // ---- f(1): single extra evaluation on VALU, reusing resident E fragments
  const float a0l = p[lane], b0l = p[32 + lane];
  const float h1 = fsigmoid(a0l * (1.0f - b0l));     // lane i holds h(1)[i]
  float ps0 = 0.0f, ps1 = 0.0f;
#pragma unroll
  for (int j = 0; j < 8; ++j) {
    const float hx = lane_gather(h1, 4 * j + 2 * s);
    const float hy = lane_gather(h1, 4 * j + 2 * s + 1);
    ps0 += ea[0][j].x * hx + ea[0][j].y * hy;
    ps1 += ea[1][j].x * hx + ea[1][j].y * hy;
  }
  ps0 += lane_gather(ps0, lane ^ 16);
  ps1 += lane_gather(ps1, lane ^ 16);
  const float g0 = fsigmoid(ps0 - c1m0);
  const float g1 = fsigmoid(ps1 - c1m1);
  const float q = lane_gather(w2l, r) * g0 + lane_gather(w2l, 16 + r) * g1;
  const float f1 = 0.5f * wave_sum32(q, lane) + NITS_MONO * 1.0f;  // rows counted twice

  // ---- two-level 32-way section search (identical grid to 10-step bisection)
  const float yv = z[sample];
  float thr = 0.0f;
  int k0 = 0, j0 = 0;
  float xbase = 0.0f, xstep = 1.0f / 32.0f;

  for (int round = 0; round < 2; ++round) {
    const float xA = xbase + (float)r * xstep;          // columns 0..15
    const float xB = xbase + (float)(16 + r) * xstep;   // columns 16..31
    v8f acc00 = {}, acc01 = {}, acc10 = {}, acc11 = {};
#pragma unroll
    for (int j = 0; j < 8; ++j) {
      // B fragment rows 2s,2s+1 of K-step j: h[i](x_col) = sigmoid(a0[i]*(x-b0[i]))
      v2f bA, bB;
      bA.x = fsigmoid(a0g[j][0] * (xA - b0g[j][0]));
      bA.y = fsigmoid(a0g[j][1] * (xA - b0g[j][1]));
      bB.x = fsigmoid(a0g[j][0] * (xB - b0g[j][0]));
      bB.y = fsigmoid(a0g[j][1] * (xB - b0g[j][1]));
      acc00 = __builtin_amdgcn_wmma_f32_16x16x4_f32(false, ea[0][j], false, bA, (short)0, acc00, false, false);
      acc01 = __builtin_amdgcn_wmma_f32_16x16x4_f32(false, ea[0][j], false, bB, (short)0, acc01, false, false);
      acc10 = __builtin_amdgcn_wmma_f32_16x16x4_f32(false, ea[1][j], false, bA, (short)0, acc10, false, false);
      acc11 = __builtin_amdgcn_wmma_f32_16x16x4_f32(false, ea[1][j], false, bB, (short)0, acc11, false, false);
    }
    // layer-1 bias + sigmoid, then layer-2 softmax dot (partial over this
    // lane's 16 rows), completed across the lane halves with one xor-16.
    float fpA = 0.0f, fpB = 0.0f;
#pragma unroll
    for (int v = 0; v < 8; ++v) {
      fpA += w2g[0][v] * fsigmoid(acc00[v] - c1g[0][v]);
      fpB += w2g[0][v] * fsigmoid(acc01[v] - c1g[0][v]);
      fpA += w2g[1][v] * fsigmoid(acc10[v] - c1g[1][v]);
      fpB += w2g[1][v] * fsigmoid(acc11[v] - c1g[1][v]);
    }
    fpA += lane_gather(fpA, lane ^ 16);
    fpB += lane_gather(fpB, lane ^ 16);
    const float fA = fpA + NITS_MONO * xA;
    const float fB = fpB + NITS_MONO * xB;

    if (round == 0) {                       // column 0 of round 0 is f(START)
      const float f0 = lane_gather(fA, 0);
      thr = f0 + yv * (f1 - f0);            // f(x) > thr  <=>  cdf(x) > y
    }
    int cand = (fA <= thr) ? r : 0;         // largest grid index with f <= thr
    if (fB <= thr) { const int c2 = 16 + r; cand = (cand > c2) ? cand : c2; }
    const int sel = wave_maxi32(cand, lane);
    if (round == 0) {
      k0 = sel;
      xbase = (float)k0 * (1.0f / 32.0f);
      xstep = 1.0f / 1024.0f;
    } else {
      j0 = sel;
    }
  }

  if (lane == 0)
    out[sample] = (float)(k0 * 32 + j0 + 1) * (1.0f / 1024.0f);  // == bisection "high"
}

extern "C" void kernel_launch(void* const* d_in, const int* in_sizes, int n_in,
                              void* d_out, int out_size, void* d_ws, size_t ws_size,
                              hipStream_t stream) {
  (void)n_in; (void)out_size; (void)d_ws; (void)ws_size;
  const float* z      = (const float*)d_in[0];
  const float* params = (const float*)d_in[1];
  float* out = (float*)d_out;
  const int n = in_sizes[0];                       // 262144 samples
  const int blocks = (n + WAVES_PER_BLOCK - 1) / WAVES_PER_BLOCK;
  NITS_invcdf_wmma_kernel<<<blocks, 32 * WAVES_PER_BLOCK, 0, stream>>>(z, params, out, n);
}